// GroupKernel_7902739824810
// MI455X (gfx1250) — compile-verified
//
#include <hip/hip_runtime.h>
#include <hip/hip_bf16.h>
#include <math.h>

// ---------------------------------------------------------------------------
// out[o,i,ii,j,x,y] = sum_t Bt[(i,j,x,y), t] * signal[(o,ii), t],  t=(d,h,w)
// GEMM:  C[CH=16384, S=3136] = A[CH, KD=392] x Bt^T   (Bt stored [S, KD])
// Memory-bound: ~231MB HBM -> ~10us floor @ 23.3 TB/s. fp32 WMMA 16x16x4.
// ---------------------------------------------------------------------------
#define O_ 128
#define I_ 128
#define N_ 8
#define K_ 7
#define CH (O_ * I_)            // 16384 GEMM rows (channels)
#define S_ (N_ * N_ * K_ * K_)  // 3136  GEMM cols (i,j,x,y)
#define KD (N_ * K_ * K_)       // 392   GEMM inner dim (d,h,w)

#define TWO_PI_F 6.283185307179586f

typedef float v2f __attribute__((ext_vector_type(2)));
typedef float v8f __attribute__((ext_vector_type(8)));
typedef int   v4i __attribute__((ext_vector_type(4)));

#define GLOBAL_AS __attribute__((address_space(1)))
#define LDS_AS    __attribute__((address_space(3)))

#if __has_builtin(__builtin_amdgcn_global_load_async_to_lds_b128)
#define HAS_ASYNC_LDS 1
#else
#define HAS_ASYNC_LDS 0
#endif

__device__ __forceinline__ void copy16_g2lds(const float* gsrc, float* ldst) {
#if HAS_ASYNC_LDS
    __builtin_amdgcn_global_load_async_to_lds_b128(
        (GLOBAL_AS v4i*)(v4i*)gsrc, (LDS_AS v4i*)(v4i*)ldst, 0, 0);
#else
    *(float4*)ldst = *(const float4*)gsrc;
#endif
}

__device__ __forceinline__ void wait_async_lds() {
#if HAS_ASYNC_LDS
#if __has_builtin(__builtin_amdgcn_s_wait_asynccnt)
    __builtin_amdgcn_s_wait_asynccnt(0);
#else
    asm volatile("s_wait_asynccnt 0x0" ::: "memory");
#endif
#endif
}

// -------------------------- Bt-matrix construction -------------------------

__global__ void zeroB_kernel(float4* __restrict__ b, int n4) {
    int i = blockIdx.x * blockDim.x + threadIdx.x;
    if (i < n4) b[i] = make_float4(0.f, 0.f, 0.f, 0.f);
}

__device__ __forceinline__ void prep_tap(float c, int size,
                                         int& lo, int& hi, float& fr,
                                         bool& vlo, bool& vhi) {
    float pos = (c + 1.0f) * 0.5f * (float)(size - 1);
    float flo = floorf(pos);
    fr = pos - flo;
    int lo_i = (int)flo;
    int hi_i = lo_i + 1;
    vlo = (lo_i >= 0) && (lo_i <= size - 1);
    vhi = (hi_i >= 0) && (hi_i <= size - 1);
    lo = min(max(lo_i, 0), size - 1);
    hi = min(max(hi_i, 0), size - 1);
}

__global__ void buildB_kernel(float* __restrict__ Bt) {   // Bt[S_][KD]
    int s = blockIdx.x * blockDim.x + threadIdx.x;
    if (s >= S_) return;
    int i   = s / (N_ * K_ * K_);
    int rem = s - i * (N_ * K_ * K_);
    int j   = rem / (K_ * K_);
    int xy  = rem - j * (K_ * K_);
    int x   = xy / K_;
    int y   = xy - x * K_;

    float ei = (float)i * (TWO_PI_F / (float)N_);
    float ej = (float)j * (TWO_PI_F / (float)N_);

    float inv = fmodf(-ei, TWO_PI_F); if (inv < 0.f) inv += TWO_PI_F;
    float cs = cosf(inv), sn = sinf(inv);
    float gx = -1.0f + 2.0f * (float)x / (float)(K_ - 1);
    float gy = -1.0f + 2.0f * (float)y / (float)(K_ - 1);
    float rx = cs * gx - sn * gy;
    float ry = sn * gx + cs * gy;

    float tH = fmodf(ej - ei, TWO_PI_F); if (tH < 0.f) tH += TWO_PI_F;
    float largest = TWO_PI_F * (float)(N_ - 1) / (float)N_;
    float tHn = 2.0f * tH / largest - 1.0f;

    int d0, d1, h0, h1, w0, w1; float fd, fh, fw;
    bool vd0, vd1, vh0, vh1, vw0, vw1;
    prep_tap(tHn, N_, d0, d1, fd, vd0, vd1);
    prep_tap(rx,  K_, h0, h1, fh, vh0, vh1);
    prep_tap(ry,  K_, w0, w1, fw, vw0, vw1);

    int   di[2] = {d0, d1};  float wd[2] = {1.f - fd, fd};  bool vd[2] = {vd0, vd1};
    int   hh[2] = {h0, h1};  float wh[2] = {1.f - fh, fh};  bool vh[2] = {vh0, vh1};
    int   wi[2] = {w0, w1};  float ww[2] = {1.f - fw, fw};  bool vw[2] = {vw0, vw1};

    float* row = Bt + (size_t)s * KD;   // row exclusive to this thread
    for (int a = 0; a < 2; ++a)
      for (int b = 0; b < 2; ++b)
        for (int c = 0; c < 2; ++c) {
            float tw = wd[a] * wh[b] * ww[c] *
                       ((vd[a] && vh[b] && vw[c]) ? 1.f : 0.f);
            row[di[a] * (K_ * K_) + hh[b] * K_ + wi[c]] += tw;
        }
}

// ------------------------------- WMMA GEMM ---------------------------------
// Macro-tile 128 rows x 64 cols / 256 threads; K chunked by 56; LDS stride 60
// (pad) so both A and B fragment ds_load_b64s are bank-conflict-free.

#define KC  56
#define LST 60    // LDS row stride (dwords) for both A and Bt slabs
#define RB  128   // rows per block
#define CB  64    // cols per block

__global__ __launch_bounds__(256)
void liftGemm_kernel(const float* __restrict__ A,
                     const float* __restrict__ Bt,
                     float* __restrict__ C) {
    __shared__ float Als[RB * LST];   // 30,720 B
    __shared__ float Bls[CB * LST];   // 15,360 B

    const int tid  = threadIdx.x;
    const int wave = tid >> 5;
    const int lane = tid & 31;
    const int l16  = lane & 15;
    const int lh   = lane >> 4;
    const int rowBase = blockIdx.y * RB;
    const int colBase = blockIdx.x * CB;

    v8f acc[4] = {};

    for (int kc = 0; kc < KD; kc += KC) {
        // ---- stage A chunk [128 rows x 56 K] ----
        for (int idx = tid; idx < RB * (KC / 4); idx += 256) {
            int rr = idx / (KC / 4);
            int q  = idx - rr * (KC / 4);
            copy16_g2lds(A + (size_t)(rowBase + rr) * KD + kc + q * 4,
                         &Als[rr * LST + q * 4]);
        }
        // ---- stage Bt chunk [64 cols x 56 K] (row copy, no transpose) ----
        for (int idx = tid; idx < CB * (KC / 4); idx += 256) {
            int cc = idx / (KC / 4);
            int q  = idx - cc * (KC / 4);
            copy16_g2lds(Bt + (size_t)(colBase + cc) * KD + kc + q * 4,
                         &Bls[cc * LST + q * 4]);
        }
        // prefetch next A chunk while this one is consumed
        if (kc + KC < KD && tid < RB)
            __builtin_prefetch(A + (size_t)(rowBase + tid) * KD + kc + KC, 0, 0);
        wait_async_lds();
        __syncthreads();

        // ---- WMMA: wave owns 16-row strip, 4 x (16x16) accumulators ----
        const float* ap = &Als[(wave * 16 + l16) * LST];
        for (int k0 = 0; k0 < KC; k0 += 4) {
            const int kb = k0 + 2 * lh;              // K index per ISA A layout
            v2f a = *(const v2f*)(ap + kb);          // ds_load_b64
#pragma unroll
            for (int jj = 0; jj < 4; ++jj) {
                v2f b = *(const v2f*)(&Bls[(jj * 16 + l16) * LST + kb]);
                acc[jj] = __builtin_amdgcn_wmma_f32_16x16x4_f32(
                    false, a, false, b, (short)0, acc[jj], false, false);
            }
        }
        __syncthreads();
    }

    // ---- epilogue: transpose 16x16 tiles through per-wave LDS scratch so
    //      each lane stores 8 contiguous floats (2x global_store_b128) ----
    float* Tw = &Als[wave * (16 * 20)];              // wave-private, 320 floats
    const int rch = rowBase + wave * 16 + l16;       // output channel row
    const int o   = rch >> 7;
    const int ii  = rch & 127;
#pragma unroll
    for (int jj = 0; jj < 4; ++jj) {
        // scatter fragment into LDS: T[row][col], row stride 20 (pad)
#pragma unroll
        for (int vv = 0; vv < 8; ++vv)
            Tw[(vv + 8 * lh) * 20 + l16] = acc[jj][vv];
        // gather one row-half per lane (same-wave LDS ops are in-order)
        float4 v0 = *(float4*)&Tw[l16 * 20 + lh * 8];
        float4 v1 = *(float4*)&Tw[l16 * 20 + lh * 8 + 4];
        const int c0 = colBase + jj * 16 + lh * 8;   // 8-col span: single i-group
        const int gi = c0 / KD;
        const int s2 = c0 - gi * KD;
        const size_t addr = ((size_t)(o * N_ + gi) * I_ + ii) * (size_t)KD + s2;
        *(float4*)&C[addr]     = v0;
        *(float4*)&C[addr + 4] = v1;
    }
}

// ------------------------------- launcher ----------------------------------

extern "C" void kernel_launch(void* const* d_in, const int* in_sizes, int n_in,
                              void* d_out, int out_size, void* d_ws, size_t ws_size,
                              hipStream_t stream) {
    const float* weight = (const float*)d_in[0];   // [128,128,8,7,7] fp32
    float* out = (float*)d_out;                    // [128,8,128,8,7,7] fp32
    float* Bt  = (float*)d_ws;                     // Bt [3136, 392] = 4.9 MB

    const int n4 = (KD * S_) / 4;
    zeroB_kernel<<<(n4 + 255) / 256, 256, 0, stream>>>((float4*)Bt, n4);
    buildB_kernel<<<(S_ + 255) / 256, 256, 0, stream>>>(Bt);

    dim3 grid(S_ / CB, CH / RB);   // (49, 128)
    liftGemm_kernel<<<grid, 256, 0, stream>>>(weight, Bt, out);
}